// GNNlayer_42898133353507
// MI455X (gfx1250) — compile-verified
//
#include <hip/hip_runtime.h>

typedef __attribute__((ext_vector_type(2))) float v2f;
typedef __attribute__((ext_vector_type(8))) float v8f;

constexpr int Nn   = 50000;
constexpr int Ee   = 800000;
constexpr int Fdim = 128;
constexpr int Hh   = 8;
constexpr int Fh   = 512;   // ffn hidden
constexpr float NEG = 0.2f;
constexpr float EPS = 1e-6f;

__device__ __forceinline__ v8f wmma_f32(v2f a, v2f b, v8f c) {
  // D = A(16x4,f32) * B(4x16,f32) + C(16x16,f32)
  return __builtin_amdgcn_wmma_f32_16x16x4_f32(false, a, false, b, (short)0, c,
                                               false, false);
}

// LayerNorm one row of 128 with one wave (lane handles 4 elems).
// std is unbiased (ddof=1), eps added to std -- matches reference.
__device__ __forceinline__ void ln_row_g(const float* __restrict__ row,
                                         float* __restrict__ dst,
                                         const float* __restrict__ ga,
                                         const float* __restrict__ gb,
                                         int lane) {
  float x0 = row[lane*4+0], x1 = row[lane*4+1];
  float x2 = row[lane*4+2], x3 = row[lane*4+3];
  float s = x0 + x1 + x2 + x3;
  float q = x0*x0 + x1*x1 + x2*x2 + x3*x3;
#pragma unroll
  for (int off = 1; off < 32; off <<= 1) {
    s += __shfl_xor(s, off, 32);
    q += __shfl_xor(q, off, 32);
  }
  float m   = s * (1.0f / 128.0f);
  float var = (q - 128.0f * m * m) * (1.0f / 127.0f);
  var = var > 0.0f ? var : 0.0f;
  float inv = 1.0f / (sqrtf(var) + EPS);
  dst[lane*4+0] = ga[lane*4+0] * (x0 - m) * inv + gb[lane*4+0];
  dst[lane*4+1] = ga[lane*4+1] * (x1 - m) * inv + gb[lane*4+1];
  dst[lane*4+2] = ga[lane*4+2] * (x2 - m) * inv + gb[lane*4+2];
  dst[lane*4+3] = ga[lane*4+3] * (x3 - m) * inv + gb[lane*4+3];
}

// ---------------- Kernel 1: node path -----------------------------------
// LN(ent) -> h (LDS). Wave w computes the 16x16 output tile for head w of
// the three GEMMs h@W_head, h@W_tail, h@W_ent (A-frag shared by 3 WMMAs).
// Fuses tanh + attn-vector cross-lane reductions and the ft store.
__global__ void __launch_bounds__(256) k_node(
    const float* __restrict__ ent,
    const float* __restrict__ Wh, const float* __restrict__ Wt,
    const float* __restrict__ We,
    const float* __restrict__ ah, const float* __restrict__ at,
    const float* __restrict__ lna, const float* __restrict__ lnb,
    float* __restrict__ eh, float* __restrict__ et, float* __restrict__ ft) {
  __shared__ float hs[16][132];
  int base = blockIdx.x * 16;
  int wave = threadIdx.x >> 5, lane = threadIdx.x & 31;
  for (int r = wave; r < 16; r += 8)
    ln_row_g(ent + (size_t)(base + r) * Fdim, hs[r], lna, lnb, lane);
  __syncthreads();

  const int w = wave;                 // head index == output column tile
  const int m = lane & 15;
  const int koff = (lane >> 4) << 1;  // 0 (lanes 0-15) or 2 (lanes 16-31)
  v8f ch = {}, ctl = {}, ce = {};
  const float* whc = Wh + w * 16 + m;
  const float* wtc = Wt + w * 16 + m;
  const float* wec = We + w * 16 + m;
  for (int k0 = 0; k0 < Fdim; k0 += 4) {
    int k = k0 + koff;
    float2 av = *(const float2*)(&hs[m][k]);
    v2f a; a.x = av.x; a.y = av.y;
    v2f bh; bh.x = whc[(size_t)k * Fdim]; bh.y = whc[(size_t)(k + 1) * Fdim];
    v2f bt; bt.x = wtc[(size_t)k * Fdim]; bt.y = wtc[(size_t)(k + 1) * Fdim];
    v2f be; be.x = wec[(size_t)k * Fdim]; be.y = wec[(size_t)(k + 1) * Fdim];
    ch  = wmma_f32(a, bh, ch);
    ctl = wmma_f32(a, bt, ctl);
    ce  = wmma_f32(a, be, ce);
  }
  float wah = ah[w * 16 + m];
  float wat = at[w * 16 + m];
#pragma unroll
  for (int r = 0; r < 8; ++r) {
    int node = base + r + ((lane >> 4) << 3);
    float ph = tanhf(ch[r])  * wah;
    float pt = tanhf(ctl[r]) * wat;
#pragma unroll
    for (int off = 1; off < 16; off <<= 1) {  // reduce within 16-lane halves
      ph += __shfl_xor(ph, off, 32);
      pt += __shfl_xor(pt, off, 32);
    }
    if (m == 0) { eh[node * Hh + w] = ph; et[node * Hh + w] = pt; }
    ft[(size_t)node * Fdim + w * 16 + m] = ce[r];  // no tanh on W_ent path
  }
}

// ---------------- Kernel 2: relation path (64 rows / block) -------------
// Each B-fragment pair is reused across 4 row-tile accumulators: B-load
// traffic per WMMA drops 4x vs a 16-row block (W_rel refetch 3.2GB->0.8GB).
__global__ void __launch_bounds__(256) k_rel(
    const float* __restrict__ relv,
    const float* __restrict__ Wr, const float* __restrict__ ar,
    const float* __restrict__ lna, const float* __restrict__ lnb,
    float* __restrict__ er) {
  __shared__ float hs[64][132];
  int base = blockIdx.x * 64;
  int wave = threadIdx.x >> 5, lane = threadIdx.x & 31;
  for (int r = wave; r < 64; r += 8)
    ln_row_g(relv + (size_t)(base + r) * Fdim, hs[r], lna, lnb, lane);
  __syncthreads();

  const int w = wave;
  const int m = lane & 15;
  const int koff = (lane >> 4) << 1;
  v8f cc[4] = {};
  const float* wrc = Wr + w * 16 + m;
  for (int k0 = 0; k0 < Fdim; k0 += 4) {
    int k = k0 + koff;
    v2f b; b.x = wrc[(size_t)k * Fdim]; b.y = wrc[(size_t)(k + 1) * Fdim];
#pragma unroll
    for (int t = 0; t < 4; ++t) {
      float2 av = *(const float2*)(&hs[t * 16 + m][k]);
      v2f a; a.x = av.x; a.y = av.y;
      cc[t] = wmma_f32(a, b, cc[t]);
    }
  }
  float war = ar[w * 16 + m];
#pragma unroll
  for (int t = 0; t < 4; ++t) {
#pragma unroll
    for (int r = 0; r < 8; ++r) {
      int e = base + t * 16 + r + ((lane >> 4) << 3);
      float p = tanhf(cc[t][r]) * war;
#pragma unroll
      for (int off = 1; off < 16; off <<= 1) p += __shfl_xor(p, off, 32);
      if (m == 0) er[(size_t)e * Hh + w] = p;
    }
  }
}

// ---------------- Kernel 3: init scratch --------------------------------
__global__ void k_init(unsigned* __restrict__ mx, float* __restrict__ den,
                       float* __restrict__ agg, int nh, int nf) {
  int i = blockIdx.x * blockDim.x + threadIdx.x;
  if (i < nh) { mx[i] = 0u; den[i] = 0.0f; }
  if (i < nf) agg[i] = 0.0f;
}

// ---------------- Kernel 4: edge logits + segment max -------------------
__global__ void k_logits(const float* __restrict__ eh, const float* __restrict__ et,
                         const float* __restrict__ er,
                         const int* __restrict__ esrc, const int* __restrict__ edst,
                         float* __restrict__ eb, unsigned* __restrict__ mx,
                         int total) {
  int i = blockIdx.x * blockDim.x + threadIdx.x;
  if (i >= total) return;
  int ei = i >> 3, h = i & 7;
  int s = esrc[ei], d = edst[ei];
  float v = eh[s * Hh + h] + et[d * Hh + h] + er[i];
  v = v >= 0.0f ? v : NEG * v;
  eb[i] = v;
  // order-preserving key for unsigned atomic max over floats
  unsigned bits = __float_as_uint(v);
  unsigned key  = (bits & 0x80000000u) ? ~bits : (bits | 0x80000000u);
  atomicMax(&mx[d * Hh + h], key);
}

// ---------------- Kernel 5: exp + segment sum ---------------------------
__global__ void k_soft(float* __restrict__ eb, const unsigned* __restrict__ mx,
                       float* __restrict__ den, const int* __restrict__ edst,
                       int total) {
  int i = blockIdx.x * blockDim.x + threadIdx.x;
  if (i >= total) return;
  int ei = i >> 3, h = i & 7;
  int d = edst[ei];
  unsigned key  = mx[d * Hh + h];
  unsigned bits = (key & 0x80000000u) ? (key ^ 0x80000000u) : ~key;
  float ex = expf(eb[i] - __uint_as_float(bits));
  eb[i] = ex;
  atomicAdd(&den[d * Hh + h], ex);
}

// ---------------- Kernel 6: weighted message scatter --------------------
// One wave per edge; lane handles 4 contiguous features (one head / lane).
__global__ void __launch_bounds__(256) k_scatter(
    const float* __restrict__ eb, const float* __restrict__ den,
    const float* __restrict__ ft, const int* __restrict__ esrc,
    const int* __restrict__ edst, float* __restrict__ agg) {
  int idx = blockIdx.x * 256 + threadIdx.x;
  int ei = idx >> 5;
  if (ei >= Ee) return;
  int lane = idx & 31;
  int f0 = lane * 4;
  int h  = f0 >> 4;
  int s = esrc[ei], d = edst[ei];
  float a = eb[(size_t)ei * Hh + h] / den[d * Hh + h];
  float4 v = *(const float4*)(ft + (size_t)s * Fdim + f0);
  float* o = agg + (size_t)d * Fdim + f0;
  atomicAdd(o + 0, v.x * a);
  atomicAdd(o + 1, v.y * a);
  atomicAdd(o + 2, v.z * a);
  atomicAdd(o + 3, v.w * a);
}

// ---------------- Kernel 7: residual + pre-norm FFN (fused) -------------
__global__ void __launch_bounds__(256) k_ffn(
    const float* __restrict__ ent, const float* __restrict__ agg,
    const float* __restrict__ fna, const float* __restrict__ fnb,
    const float* __restrict__ W1, const float* __restrict__ b1,
    const float* __restrict__ W2, const float* __restrict__ b2,
    float* __restrict__ out) {
  __shared__ float rs[16][132];
  __shared__ float zs[16][132];
  __shared__ float hid[16][516];
  int base = blockIdx.x * 16;
  int wave = threadIdx.x >> 5, lane = threadIdx.x & 31;

  // rst = ent + agg ; z = LN(rst)
  for (int r = wave; r < 16; r += 8) {
    const float* erow = ent + (size_t)(base + r) * Fdim;
    const float* grow = agg + (size_t)(base + r) * Fdim;
    float x0 = erow[lane*4+0] + grow[lane*4+0];
    float x1 = erow[lane*4+1] + grow[lane*4+1];
    float x2 = erow[lane*4+2] + grow[lane*4+2];
    float x3 = erow[lane*4+3] + grow[lane*4+3];
    rs[r][lane*4+0] = x0; rs[r][lane*4+1] = x1;
    rs[r][lane*4+2] = x2; rs[r][lane*4+3] = x3;
    float s = x0 + x1 + x2 + x3;
    float q = x0*x0 + x1*x1 + x2*x2 + x3*x3;
#pragma unroll
    for (int off = 1; off < 32; off <<= 1) {
      s += __shfl_xor(s, off, 32);
      q += __shfl_xor(q, off, 32);
    }
    float m   = s * (1.0f / 128.0f);
    float var = (q - 128.0f * m * m) * (1.0f / 127.0f);
    var = var > 0.0f ? var : 0.0f;
    float inv = 1.0f / (sqrtf(var) + EPS);
    zs[r][lane*4+0] = fna[lane*4+0] * (x0 - m) * inv + fnb[lane*4+0];
    zs[r][lane*4+1] = fna[lane*4+1] * (x1 - m) * inv + fnb[lane*4+1];
    zs[r][lane*4+2] = fna[lane*4+2] * (x2 - m) * inv + fnb[lane*4+2];
    zs[r][lane*4+3] = fna[lane*4+3] * (x3 - m) * inv + fnb[lane*4+3];
  }
  __syncthreads();

  const int m = lane & 15;
  const int koff = (lane >> 4) << 1;

  // hidden = relu(z @ W1 + b1)  (16 x 512)
  // Wave covers 4 column tiles; A-fragment shared across the 4 WMMAs.
  {
    v8f cc[4] = {};
    for (int k0 = 0; k0 < Fdim; k0 += 4) {
      int k = k0 + koff;
      float2 av = *(const float2*)(&zs[m][k]);
      v2f a; a.x = av.x; a.y = av.y;
#pragma unroll
      for (int ct = 0; ct < 4; ++ct) {
        int col = wave * 64 + ct * 16;
        v2f b; b.x = W1[(size_t)k * Fh + col + m];
        b.y = W1[(size_t)(k + 1) * Fh + col + m];
        cc[ct] = wmma_f32(a, b, cc[ct]);
      }
    }
#pragma unroll
    for (int ct = 0; ct < 4; ++ct) {
      int col = wave * 64 + ct * 16;
      float bb = b1[col + m];
#pragma unroll
      for (int r = 0; r < 8; ++r) {
        float v = cc[ct][r] + bb;
        hid[r + ((lane >> 4) << 3)][col + m] = v > 0.0f ? v : 0.0f;
      }
    }
  }
  __syncthreads();

  // out = rst + hidden @ W2 + b2  (16 x 128), wave covers column tile `wave`
  {
    int col = wave * 16;
    v8f c = {};
    for (int k0 = 0; k0 < Fh; k0 += 4) {
      int k = k0 + koff;
      float2 av = *(const float2*)(&hid[m][k]);
      v2f a; a.x = av.x; a.y = av.y;
      v2f b; b.x = W2[(size_t)k * Fdim + col + m];
      b.y = W2[(size_t)(k + 1) * Fdim + col + m];
      c = wmma_f32(a, b, c);
    }
    float bb = b2[col + m];
#pragma unroll
    for (int r = 0; r < 8; ++r) {
      int row = r + ((lane >> 4) << 3);
      out[(size_t)(base + row) * Fdim + col + m] = rs[row][col + m] + c[r] + bb;
    }
  }
}

extern "C" void kernel_launch(void* const* d_in, const int* in_sizes, int n_in,
                              void* d_out, int out_size, void* d_ws, size_t ws_size,
                              hipStream_t stream) {
  (void)in_sizes; (void)n_in; (void)out_size; (void)ws_size;
  const float* ent  = (const float*)d_in[0];
  const float* relv = (const float*)d_in[1];
  const int*   esrc = (const int*)d_in[2];
  const int*   edst = (const int*)d_in[3];
  const float* Wh   = (const float*)d_in[4];
  const float* Wt   = (const float*)d_in[5];
  const float* We   = (const float*)d_in[6];
  const float* Wr   = (const float*)d_in[7];
  const float* ah   = (const float*)d_in[8];
  const float* at   = (const float*)d_in[9];
  const float* ar   = (const float*)d_in[10];
  const float* lea  = (const float*)d_in[11];
  const float* leb  = (const float*)d_in[12];
  const float* lra  = (const float*)d_in[13];
  const float* lrb  = (const float*)d_in[14];
  const float* W1   = (const float*)d_in[15];
  const float* b1   = (const float*)d_in[16];
  const float* W2   = (const float*)d_in[17];
  const float* b2   = (const float*)d_in[18];
  const float* fna  = (const float*)d_in[19];
  const float* fnb  = (const float*)d_in[20];

  float* ws = (float*)d_ws;
  float* eh  = ws;  ws += (size_t)Nn * Hh;
  float* et  = ws;  ws += (size_t)Nn * Hh;
  float* ft  = ws;  ws += (size_t)Nn * Fdim;
  float* er  = ws;  ws += (size_t)Ee * Hh;
  float* eb  = ws;  ws += (size_t)Ee * Hh;
  unsigned* mx = (unsigned*)ws; ws += (size_t)Nn * Hh;
  float* den = ws;  ws += (size_t)Nn * Hh;
  float* agg = ws;  ws += (size_t)Nn * Fdim;

  dim3 blk(256);
  k_node<<<Nn / 16, blk, 0, stream>>>(ent, Wh, Wt, We, ah, at, lea, leb,
                                      eh, et, ft);
  k_rel<<<Ee / 64, blk, 0, stream>>>(relv, Wr, ar, lra, lrb, er);
  k_init<<<(Nn * Fdim + 255) / 256, blk, 0, stream>>>(mx, den, agg,
                                                      Nn * Hh, Nn * Fdim);
  k_logits<<<(Ee * Hh + 255) / 256, blk, 0, stream>>>(eh, et, er, esrc, edst,
                                                      eb, mx, Ee * Hh);
  k_soft<<<(Ee * Hh + 255) / 256, blk, 0, stream>>>(eb, mx, den, edst, Ee * Hh);
  k_scatter<<<(Ee * 32) / 256, blk, 0, stream>>>(eb, den, ft, esrc, edst, agg);
  k_ffn<<<Nn / 16, blk, 0, stream>>>(ent, agg, fna, fnb, W1, b1, W2, b2,
                                     (float*)d_out);
}